// RSC_81733227642889
// MI455X (gfx1250) — compile-verified
//
#include <hip/hip_runtime.h>
#include <hip/hip_bf16.h>

typedef __attribute__((ext_vector_type(2))) float v2f;
typedef __attribute__((ext_vector_type(4))) float v4f;
typedef __attribute__((ext_vector_type(8))) float v8f;

#define B_DIM     32
#define C_DIM     2048
#define K_CLASSES 400
#define SPATIAL   392          // T*H*W = 8*7*7
#define KSEL      1576u        // 0-based order-statistic rank (see derivation)
#define RETAIN_P  0.77f

// ---------------------------------------------------------------------------
// Phase 1a: G[b][c] = W[c, labels[b]]  computed as W(2048x400) x OneHot(400x32)
// with V_WMMA_F32_16X16X4_F32. One wave32 per 16x16 output tile (EXEC all-1s).
// One-hot entries are exact 0.0/1.0, so the WMMA result is the exact gather.
// ---------------------------------------------------------------------------
__global__ __launch_bounds__(32)
void RSC_gather_wmma(const float* __restrict__ W,
                     const long long* __restrict__ labels,
                     float* __restrict__ G /* [32][2048] */) {
  const int cbase = blockIdx.x * 16;    // 128 tiles over C (rows of W)
  const int bbase = blockIdx.y * 16;    // 2 tiles over B (one-hot columns)
  const int lane  = threadIdx.x;        // wave32
  const int mn    = lane & 15;          // A row (M) and B col (N) for this lane
  const int kadd  = (lane >> 4) << 1;   // lanes 0-15 hold K+0/K+1, lanes 16-31 K+2/K+3

  const long long  lab  = labels[bbase + mn];
  const float* __restrict__ arow = W + (size_t)(cbase + mn) * K_CLASSES;

  v8f acc = {};
  for (int k0 = 0; k0 < K_CLASSES; k0 += 4) {   // 400 = 100 * 4
    v2f a, b;
    a.x = arow[k0 + kadd];
    a.y = arow[k0 + kadd + 1];
    b.x = (lab == (long long)(k0 + kadd))     ? 1.0f : 0.0f;
    b.y = (lab == (long long)(k0 + kadd + 1)) ? 1.0f : 0.0f;
    acc = __builtin_amdgcn_wmma_f32_16x16x4_f32(false, a, false, b,
                                                (short)0, acc, false, false);
  }

  // D layout: VGPR r -> M = r (lanes 0-15) / r+8 (lanes 16-31), N = lane&15
  const int mofs = (lane >> 4) << 3;
#pragma unroll
  for (int r = 0; r < 8; ++r) {
    G[(size_t)(bbase + mn) * C_DIM + (size_t)(cbase + r + mofs)] = acc[r];
  }
}

// ---------------------------------------------------------------------------
// Phase 1b: per-sample exact radix-select of the 1576-th order statistic of
// g_c = G[b][c]/392, then emit the per-(b,c) multiplier table:
//   unchanged -> 1.0 ; g_c > thr -> 0.0 ; else 1/0.77
// Keys are the standard order-preserving uint32 mapping of f32.
// ---------------------------------------------------------------------------
__global__ __launch_bounds__(256)
void RSC_select(const float* __restrict__ G,
                const int* __restrict__ unchanged,
                float* __restrict__ mult /* [32][2048] */) {
  const int b = blockIdx.x;
  const int t = threadIdx.x;
  __shared__ unsigned int hist[256];
  __shared__ unsigned int s_prefix, s_k;

  unsigned int key[8];
#pragma unroll
  for (int j = 0; j < 8; ++j) {
    const int c = t + 256 * j;
    const float g = G[(size_t)b * C_DIM + c] / 392.0f;   // mean-grad scaling
    const unsigned int bits = __float_as_uint(g);
    key[j] = bits ^ ((bits & 0x80000000u) ? 0xFFFFFFFFu : 0x80000000u);
  }
  if (t == 0) { s_prefix = 0u; s_k = KSEL; }
  __syncthreads();

  for (int pass = 0; pass < 4; ++pass) {
    const int shift = 24 - 8 * pass;
    hist[t] = 0u;
    __syncthreads();
    const unsigned int pmask = (pass == 0) ? 0u : (0xFFFFFFFFu << (shift + 8));
    const unsigned int pref  = s_prefix;
#pragma unroll
    for (int j = 0; j < 8; ++j) {
      if ((key[j] & pmask) == pref)
        atomicAdd(&hist[(key[j] >> shift) & 0xFFu], 1u);
    }
    __syncthreads();
    if (t == 0) {
      unsigned int k = s_k, cum = 0u;
      int d = 0;
      for (; d < 255; ++d) {            // bin 255 is the catch-all
        const unsigned int h = hist[d];
        if (cum + h > k) break;
        cum += h;
      }
      s_k = k - cum;
      s_prefix |= ((unsigned int)d) << shift;
    }
    __syncthreads();
  }

  const unsigned int thrkey = s_prefix;         // key of the exact threshold value
  const bool  keepall   = unchanged[b] != 0;
  const float passscale = 1.0f / RETAIN_P;
#pragma unroll
  for (int j = 0; j < 8; ++j) {
    const int c = t + 256 * j;
    const float m = keepall ? 1.0f : ((key[j] > thrkey) ? 0.0f : passscale);
    mult[(size_t)b * C_DIM + c] = m;
  }
}

// ---------------------------------------------------------------------------
// Phase 2: out = features * mult[row]. Pure streaming: 206 MB total -> ~9 us
// at 23.3 TB/s. b128 non-temporal loads/stores; mult (256 KB) stays cached.
// 392 floats/row = 98 float4s/row, so every float4 is within one row.
// ---------------------------------------------------------------------------
__global__ __launch_bounds__(256)
void RSC_apply(const v4f* __restrict__ f, const float* __restrict__ mult,
               v4f* __restrict__ out, unsigned int n4) {
  const unsigned int i = blockIdx.x * 256u + threadIdx.x;
  if (i >= n4) return;
  const unsigned int row = i / 98u;             // (b*C + c) row index
  const float m = mult[row];
  v4f v = __builtin_nontemporal_load(f + i);
  v.x *= m; v.y *= m; v.z *= m; v.w *= m;
  __builtin_nontemporal_store(v, out + i);
}

// ---------------------------------------------------------------------------
extern "C" void kernel_launch(void* const* d_in, const int* in_sizes, int n_in,
                              void* d_out, int out_size, void* d_ws, size_t ws_size,
                              hipStream_t stream) {
  const float*      features  = (const float*)d_in[0];      // [32,2048,8,7,7] f32
  const float*      W         = (const float*)d_in[1];      // [2048,400] f32
  const long long*  labels    = (const long long*)d_in[2];  // [32] i64
  const int*        unchanged = (const int*)d_in[3];        // [32] i32
  float*            out       = (float*)d_out;              // [32,2048,8,7,7] f32

  float* G    = (float*)d_ws;                // 32*2048 floats = 256 KB
  float* mult = G + (size_t)B_DIM * C_DIM;   // 32*2048 floats = 256 KB

  // Phase 1a: WMMA one-hot GEMM gather of W columns.
  RSC_gather_wmma<<<dim3(C_DIM / 16, B_DIM / 16), 32, 0, stream>>>(W, labels, G);

  // Phase 1b: exact per-sample radix-select threshold + multiplier table.
  RSC_select<<<B_DIM, 256, 0, stream>>>(G, unchanged, mult);

  // Phase 2: bandwidth-bound streaming apply.
  const unsigned int n4 = (unsigned int)(out_size / 4);
  RSC_apply<<<(n4 + 255u) / 256u, 256, 0, stream>>>((const v4f*)features, mult,
                                                    (v4f*)out, n4);
}